// StaticExpert_21543555957369
// MI455X (gfx1250) — compile-verified
//
#include <hip/hip_runtime.h>
#include <math.h>

typedef __attribute__((ext_vector_type(2))) float v2f;
typedef __attribute__((ext_vector_type(8))) float v8f;

// ---------------------------------------------------------------------------
// Dense projection: OUT[nrows x NCOLS] = X[nrows x K] @ W[K x NCOLS]  (fp32)
// Uses V_WMMA_F32_16X16X4_F32. One wave computes a 16-row slab (NCOLS/16
// 16x16 tiles). W is staged transposed in LDS (stride K+2 to keep 8B
// alignment and break bank conflicts).
// fp32 A 16x4 layout (ISA 7.12.2): lanes 0-15 hold K=kb,kb+1; lanes 16-31
// hold K=kb+2,kb+3; M = lane&15. B 4x16 mirrors it with N = lane&15.
// C/D 16x16: VGPR r -> row r (lanes 0-15) / row r+8 (lanes 16-31).
// ---------------------------------------------------------------------------
template <int K, int NCOLS>
__global__ __launch_bounds__(256) void gat_gemm_wmma(
    const float* __restrict__ X, const float* __restrict__ W,
    float* __restrict__ OUT, int nrows) {
  constexpr int NT = NCOLS / 16;
  constexpr int KP = K + 2;
  __shared__ float sWt[NCOLS * KP];
  for (int i = threadIdx.x; i < K * NCOLS; i += 256) {
    int k = i / NCOLS, n = i % NCOLS;
    sWt[n * KP + k] = W[i];
  }
  __syncthreads();

  const int wave = threadIdx.x >> 5;
  const int lane = threadIdx.x & 31;
  const int m16 = lane & 15;
  const int kh = lane >> 4;  // which K-half of the fragment this lane holds
  const int rowBase = (blockIdx.x * 8 + wave) * 16;
  int row = rowBase + m16;
  if (row >= nrows) row = nrows - 1;  // clamp loads; EXEC stays all-ones
  const float* __restrict__ xrow = X + (size_t)row * K;

  v8f acc[NT];
#pragma unroll
  for (int t = 0; t < NT; ++t)
    acc[t] = (v8f){0.f, 0.f, 0.f, 0.f, 0.f, 0.f, 0.f, 0.f};

#pragma unroll
  for (int kb = 0; kb < K; kb += 4) {
    v2f a = *(const v2f*)(xrow + kb + 2 * kh);
#pragma unroll
    for (int t = 0; t < NT; ++t) {
      const int n = t * 16 + m16;
      v2f b = *(const v2f*)(&sWt[n * KP + kb + 2 * kh]);
      acc[t] = __builtin_amdgcn_wmma_f32_16x16x4_f32(
          false, a, false, b, (short)0, acc[t], false, false);
    }
  }

#pragma unroll
  for (int r = 0; r < 8; ++r) {
    const int orow = rowBase + r + 8 * kh;
    if (orow < nrows) {
#pragma unroll
      for (int t = 0; t < NT; ++t)
        OUT[(size_t)orow * NCOLS + t * 16 + m16] = acc[t][r];
    }
  }
}

// ---------------------------------------------------------------------------
// Per-node attention scores: s[n,h] = dot(h[n,h,:], a_src[h,:]) etc.
// ---------------------------------------------------------------------------
template <int HEADS, int CH>
__global__ void gat_scores(const float* __restrict__ H,
                           const float* __restrict__ aS,
                           const float* __restrict__ aD, float* __restrict__ s,
                           float* __restrict__ d, int n) {
  int t = blockIdx.x * blockDim.x + threadIdx.x;
  if (t >= n * HEADS) return;
  int node = t / HEADS, h = t % HEADS;
  const float* hp = H + (size_t)node * HEADS * CH + h * CH;
  float sv = 0.f, dv = 0.f;
#pragma unroll
  for (int c = 0; c < CH; ++c) {
    float v = hp[c];
    sv += v * aS[h * CH + c];
    dv += v * aD[h * CH + c];
  }
  s[t] = sv;
  d[t] = dv;
}

__global__ void fill_kernel(float* __restrict__ p, float v, long long n) {
  long long t = (long long)blockIdx.x * blockDim.x + threadIdx.x;
  if (t < n) p[t] = v;
}

// Exact float atomic max via signed/unsigned integer ordering trick.
__device__ __forceinline__ void atomicMaxFloat(float* addr, float v) {
  if (v >= 0.f)
    atomicMax((int*)addr, __float_as_int(v));
  else
    atomicMin((unsigned int*)addr, __float_as_uint(v));
}

__device__ __forceinline__ void edge_endpoints(int e, int nReal,
                                               const int* __restrict__ srcIdx,
                                               const int* __restrict__ dstIdx,
                                               int& si, int& di) {
  if (e < nReal) {
    si = srcIdx[e];
    di = dstIdx[e];
  } else {
    si = di = e - nReal;  // appended self-loop
  }
}

template <int HEADS>
__global__ void gat_edge_max(const int* __restrict__ srcIdx,
                             const int* __restrict__ dstIdx,
                             const float* __restrict__ s,
                             const float* __restrict__ d, float* __restrict__ m,
                             int nReal, int nTot) {
  int e = blockIdx.x * blockDim.x + threadIdx.x;
  if (e >= nTot) return;
  int si, di;
  edge_endpoints(e, nReal, srcIdx, dstIdx, si, di);
#pragma unroll
  for (int h = 0; h < HEADS; ++h) {
    float v = s[si * HEADS + h] + d[di * HEADS + h];
    v = v > 0.f ? v : 0.2f * v;  // leaky_relu slope 0.2
    atomicMaxFloat(&m[di * HEADS + h], v);
  }
}

template <int HEADS>
__global__ void gat_edge_exp(const int* __restrict__ srcIdx,
                             const int* __restrict__ dstIdx,
                             const float* __restrict__ s,
                             const float* __restrict__ d,
                             const float* __restrict__ m,
                             float* __restrict__ ebuf, float* __restrict__ z,
                             int nReal, int nTot) {
  int e = blockIdx.x * blockDim.x + threadIdx.x;
  if (e >= nTot) return;
  int si, di;
  edge_endpoints(e, nReal, srcIdx, dstIdx, si, di);
#pragma unroll
  for (int h = 0; h < HEADS; ++h) {
    float v = s[si * HEADS + h] + d[di * HEADS + h];
    v = v > 0.f ? v : 0.2f * v;
    float ex = __expf(v - m[di * HEADS + h]);
    ebuf[(size_t)e * HEADS + h] = ex;
    atomicAdd(&z[di * HEADS + h], ex);
  }
}

// One thread per (edge, head*channel): maximizes L2 atomic parallelism.
template <int HEADS, int CH>
__global__ void gat_edge_aggr(const int* __restrict__ srcIdx,
                              const int* __restrict__ dstIdx,
                              const float* __restrict__ ebuf,
                              const float* __restrict__ z,
                              const float* __restrict__ H,
                              float* __restrict__ agg, int nReal, int nTot) {
  constexpr int HC = HEADS * CH;
  long long t = (long long)blockIdx.x * blockDim.x + threadIdx.x;
  if (t >= (long long)nTot * HC) return;
  int e = (int)(t / HC);
  int hc = (int)(t % HC);
  int h = hc / CH;
  int si, di;
  edge_endpoints(e, nReal, srcIdx, dstIdx, si, di);
  float alpha = ebuf[(size_t)e * HEADS + h] / (z[di * HEADS + h] + 1e-16f);
  atomicAdd(&agg[(size_t)di * HC + hc], H[(size_t)si * HC + hc] * alpha);
}

__global__ void gat_elu_bias(float* __restrict__ x,
                             const float* __restrict__ bias, long long total,
                             int hcMask) {
  long long t = (long long)blockIdx.x * blockDim.x + threadIdx.x;
  if (t >= total) return;
  float v = x[t] + bias[(int)t & hcMask];
  x[t] = v > 0.f ? v : __expf(v) - 1.f;  // ELU(alpha=1)
}

__global__ void gat_add_bias(float* __restrict__ x,
                             const float* __restrict__ bias, long long total,
                             int hcMask) {
  long long t = (long long)blockIdx.x * blockDim.x + threadIdx.x;
  if (t >= total) return;
  x[t] += bias[(int)t & hcMask];
}

// ---------------------------------------------------------------------------
extern "C" void kernel_launch(void* const* d_in, const int* in_sizes, int n_in,
                              void* d_out, int out_size, void* d_ws,
                              size_t ws_size, hipStream_t stream) {
  const float* x = (const float*)d_in[0];
  const int* ei = (const int*)d_in[1];
  const float* W1 = (const float*)d_in[2];
  const float* aS1 = (const float*)d_in[3];
  const float* aD1 = (const float*)d_in[4];
  const float* b1 = (const float*)d_in[5];
  const float* W2 = (const float*)d_in[6];
  const float* aS2 = (const float*)d_in[7];
  const float* aD2 = (const float*)d_in[8];
  const float* b2 = (const float*)d_in[9];
  float* out = (float*)d_out;

  const int N = in_sizes[0] / 128;    // 50000
  const int Ereal = in_sizes[1] / 2;  // 800000
  const int Etot = Ereal + N;         // + self loops
  const int* srcIdx = ei;             // edge_index[0]
  const int* dstIdx = ei + Ereal;     // edge_index[1]

  // Workspace carve-out (~38 MB of f32)
  float* p = (float*)d_ws;
  float* h1_pre = p; p += (size_t)N * 32;
  float* agg1   = p; p += (size_t)N * 32;   // becomes h1 after ELU
  float* s1     = p; p += (size_t)N * 2;
  float* d1     = p; p += (size_t)N * 2;
  float* m1     = p; p += (size_t)N * 2;
  float* z1     = p; p += (size_t)N * 2;
  float* ebuf1  = p; p += (size_t)Etot * 2;
  float* h2_pre = p; p += (size_t)N * 64;
  float* s2     = p; p += (size_t)N;
  float* d2     = p; p += (size_t)N;
  float* m2     = p; p += (size_t)N;
  float* z2     = p; p += (size_t)N;
  float* ebuf2  = p; p += (size_t)Etot;

  const int TB = 256;
  auto cdiv = [](long long a, long long b) { return (int)((a + b - 1) / b); };

  // ---------------- Layer 1 (heads=2, ch=16) ----------------
  gat_gemm_wmma<128, 32><<<cdiv(N, 128), TB, 0, stream>>>(x, W1, h1_pre, N);
  gat_scores<2, 16><<<cdiv((long long)N * 2, TB), TB, 0, stream>>>(
      h1_pre, aS1, aD1, s1, d1, N);
  fill_kernel<<<cdiv((long long)N * 2, TB), TB, 0, stream>>>(
      m1, -INFINITY, (long long)N * 2);
  fill_kernel<<<cdiv((long long)N * 2, TB), TB, 0, stream>>>(
      z1, 0.f, (long long)N * 2);
  fill_kernel<<<cdiv((long long)N * 32, TB), TB, 0, stream>>>(
      agg1, 0.f, (long long)N * 32);
  gat_edge_max<2><<<cdiv(Etot, TB), TB, 0, stream>>>(srcIdx, dstIdx, s1, d1,
                                                     m1, Ereal, Etot);
  gat_edge_exp<2><<<cdiv(Etot, TB), TB, 0, stream>>>(srcIdx, dstIdx, s1, d1,
                                                     m1, ebuf1, z1, Ereal, Etot);
  gat_edge_aggr<2, 16><<<cdiv((long long)Etot * 32, TB), TB, 0, stream>>>(
      srcIdx, dstIdx, ebuf1, z1, h1_pre, agg1, Ereal, Etot);
  gat_elu_bias<<<cdiv((long long)N * 32, TB), TB, 0, stream>>>(
      agg1, b1, (long long)N * 32, 31);

  // ---------------- Layer 2 (heads=1, ch=64) ----------------
  gat_gemm_wmma<32, 64><<<cdiv(N, 128), TB, 0, stream>>>(agg1, W2, h2_pre, N);
  gat_scores<1, 64><<<cdiv(N, TB), TB, 0, stream>>>(h2_pre, aS2, aD2, s2, d2,
                                                    N);
  fill_kernel<<<cdiv(N, TB), TB, 0, stream>>>(m2, -INFINITY, N);
  fill_kernel<<<cdiv(N, TB), TB, 0, stream>>>(z2, 0.f, N);
  fill_kernel<<<cdiv((long long)N * 64, TB), TB, 0, stream>>>(
      out, 0.f, (long long)N * 64);
  gat_edge_max<1><<<cdiv(Etot, TB), TB, 0, stream>>>(srcIdx, dstIdx, s2, d2,
                                                     m2, Ereal, Etot);
  gat_edge_exp<1><<<cdiv(Etot, TB), TB, 0, stream>>>(srcIdx, dstIdx, s2, d2,
                                                     m2, ebuf2, z2, Ereal, Etot);
  gat_edge_aggr<1, 64><<<cdiv((long long)Etot * 64, TB), TB, 0, stream>>>(
      srcIdx, dstIdx, ebuf2, z2, h2_pre, out, Ereal, Etot);
  gat_add_bias<<<cdiv((long long)N * 64, TB), TB, 0, stream>>>(
      out, b2, (long long)N * 64, 63);
}